// GCN_83356725281206
// MI455X (gfx1250) — compile-verified
//
#include <hip/hip_runtime.h>
#include <hip/hip_bf16.h>
#include <math.h>

// ---------------------------------------------------------------------------
// GCN forward, reordered:  spmm(x) @ W == spmm(x @ W)  (spmm is linear)
//   K1: y1 = X @ W1                     (WMMA f32 16x16x4, 2 M-tiles/wave)
//   K2: s1 = A @ y1                     (edge scatter, 16-wide, L2-resident)
//   K3: s2 = A @ relu(s1)               (relu fused into gather)
//   K4: out = log_softmax(s2 @ W2)      (W2 in LDS, per-row VALU)
// Workspace: y1[N*16] | s1[N*16] | s2[N*16]  = 19.2 MB
// N=100000 is divisible by 32 -> no tail handling anywhere in K1.
// ---------------------------------------------------------------------------

typedef float v2f __attribute__((ext_vector_type(2)));
typedef float v8f __attribute__((ext_vector_type(8)));

#define NN 100000
#define EE 3200000
#define IN_DIM 256
#define HID 16
#define CLS 40

__global__ void GCN_zero_f32(float* __restrict__ p, int n) {
    int i = blockIdx.x * blockDim.x + threadIdx.x;
    if (i < n) p[i] = 0.0f;
}

// One wave computes a 32x16 output slab (two 16x16 WMMA tiles sharing the
// same B fragments), accumulating over K=256 in steps of 4 with
// V_WMMA_F32_16X16X4_F32.
// A (16x4 f32) layout per ISA: lanes 0-15: (M=lane, K=k0 / k0+1) in v0/v1;
// lanes 16-31: (M=lane-16, K=k0+2 / k0+3).  -> one float2 load per lane.
// B (4x16) mirrored: v0 = B[k0+2*half][n=lid], v1 = B[k0+1+2*half][n=lid].
__global__ __launch_bounds__(128) void GCN_gemm_x_w1(
        const float* __restrict__ X, const float* __restrict__ W1,
        float* __restrict__ Y) {
    const int lane = threadIdx.x & 31;
    const int wave = threadIdx.x >> 5;
    const int slab = blockIdx.x * 4 + wave;      // 32-row slab index
    const int numSlabs = NN / 32;                // 3125, exact
    if (slab >= numSlabs) return;                // wave-uniform

    const int half = lane >> 4;                  // 0 = lanes 0-15, 1 = 16-31
    const int lid  = lane & 15;

    const int row0 = slab * 32 + lid;            // tile 0 rows
    const int row1 = row0 + 16;                  // tile 1 rows
    const float* __restrict__ a0p = X + (size_t)row0 * IN_DIM + 2 * half;
    const float* __restrict__ a1p = X + (size_t)row1 * IN_DIM + 2 * half;

    v8f acc0 = {0.f, 0.f, 0.f, 0.f, 0.f, 0.f, 0.f, 0.f};
    v8f acc1 = {0.f, 0.f, 0.f, 0.f, 0.f, 0.f, 0.f, 0.f};

    #pragma unroll 4
    for (int k0 = 0; k0 < IN_DIM; k0 += 4) {
        const float* __restrict__ bp = W1 + (size_t)(k0 + 2 * half) * HID + lid;
        v2f b;
        b.x = bp[0];
        b.y = bp[HID];
        v2f a0 = *(const v2f*)(a0p + k0);
        v2f a1 = *(const v2f*)(a1p + k0);
        // (neg_a, A, neg_b, B, c_mod, C, reuse_a, reuse_b)
        acc0 = __builtin_amdgcn_wmma_f32_16x16x4_f32(
            false, a0, false, b, (short)0, acc0, false, false);
        acc1 = __builtin_amdgcn_wmma_f32_16x16x4_f32(
            false, a1, false, b, (short)0, acc1, false, false);
    }

    // D layout: VGPR i -> row (i + 8*half), col = lid
    float* __restrict__ out0 = Y + (size_t)(slab * 32 + 8 * half) * HID + lid;
    float* __restrict__ out1 = out0 + 16 * HID;
    #pragma unroll
    for (int i = 0; i < 8; ++i) {
        out0[i * HID] = acc0[i];
        out1[i * HID] = acc1[i];
    }
}

// One thread per edge: gather 16-float src row (4x float4), scale by edge
// weight, scatter-add into dst row with native f32 global atomics.
// Rows are 64B -> gathers hit L2 (feature matrix is 6.4MB << 192MB L2).
template <bool RELU>
__global__ __launch_bounds__(256) void GCN_spmm_scatter(
        const int* __restrict__ src, const int* __restrict__ dst,
        const float* __restrict__ ew, const float* __restrict__ F,
        float* __restrict__ O) {
    int e = blockIdx.x * blockDim.x + threadIdx.x;
    if (e >= EE) return;
    int s = src[e];
    int d = dst[e];
    float w = ew[e];
    const float4* __restrict__ fr = (const float4*)(F + (size_t)s * HID);
    float* __restrict__ orow = O + (size_t)d * HID;
    #pragma unroll
    for (int q = 0; q < 4; ++q) {
        float4 v = fr[q];
        if (RELU) {
            v.x = fmaxf(v.x, 0.f); v.y = fmaxf(v.y, 0.f);
            v.z = fmaxf(v.z, 0.f); v.w = fmaxf(v.w, 0.f);
        }
        __hip_atomic_fetch_add(orow + 4 * q + 0, w * v.x,
                               __ATOMIC_RELAXED, __HIP_MEMORY_SCOPE_AGENT);
        __hip_atomic_fetch_add(orow + 4 * q + 1, w * v.y,
                               __ATOMIC_RELAXED, __HIP_MEMORY_SCOPE_AGENT);
        __hip_atomic_fetch_add(orow + 4 * q + 2, w * v.z,
                               __ATOMIC_RELAXED, __HIP_MEMORY_SCOPE_AGENT);
        __hip_atomic_fetch_add(orow + 4 * q + 3, w * v.w,
                               __ATOMIC_RELAXED, __HIP_MEMORY_SCOPE_AGENT);
    }
}

// out = log_softmax(s2 @ W2).  W2 (16x40 = 2.56KB) staged in LDS per block;
// one thread per node row: 640 FMAs + rowwise max / exp-sum / log.
__global__ __launch_bounds__(256) void GCN_head(
        const float* __restrict__ S, const float* __restrict__ W2,
        float* __restrict__ OUT) {
    __shared__ float w2s[HID * CLS];
    for (int i = threadIdx.x; i < HID * CLS; i += blockDim.x) w2s[i] = W2[i];
    __syncthreads();

    int row = blockIdx.x * blockDim.x + threadIdx.x;
    if (row >= NN) return;

    float s[HID];
    const float4* __restrict__ sr = (const float4*)(S + (size_t)row * HID);
    #pragma unroll
    for (int q = 0; q < 4; ++q) {
        float4 v = sr[q];
        s[4 * q + 0] = v.x; s[4 * q + 1] = v.y;
        s[4 * q + 2] = v.z; s[4 * q + 3] = v.w;
    }

    float acc[CLS];
    #pragma unroll
    for (int c = 0; c < CLS; ++c) acc[c] = 0.f;
    #pragma unroll
    for (int k = 0; k < HID; ++k) {
        float sk = s[k];
        const float* __restrict__ wr = &w2s[k * CLS];
        #pragma unroll
        for (int c = 0; c < CLS; ++c) acc[c] = fmaf(sk, wr[c], acc[c]);
    }

    float m = acc[0];
    #pragma unroll
    for (int c = 1; c < CLS; ++c) m = fmaxf(m, acc[c]);
    float sum = 0.f;
    #pragma unroll
    for (int c = 0; c < CLS; ++c) sum += __expf(acc[c] - m);
    float lse = m + __logf(sum);

    float* __restrict__ orow = OUT + (size_t)row * CLS;
    #pragma unroll
    for (int c = 0; c < CLS; ++c) orow[c] = acc[c] - lse;
}

extern "C" void kernel_launch(void* const* d_in, const int* in_sizes, int n_in,
                              void* d_out, int out_size, void* d_ws, size_t ws_size,
                              hipStream_t stream) {
    const float* x    = (const float*)d_in[0];   // [N, 256]
    const float* w1   = (const float*)d_in[1];   // [256, 16]
    const float* w2   = (const float*)d_in[2];   // [16, 40]
    const int*   esrc = (const int*)d_in[3];     // [E]
    const int*   edst = (const int*)d_in[4];     // [E]
    const float* ew   = (const float*)d_in[5];   // [E]
    float* out = (float*)d_out;                  // [N, 40]

    float* y1 = (float*)d_ws;                    // [N, 16]
    float* s1 = y1 + (size_t)NN * HID;           // [N, 16]
    float* s2 = s1 + (size_t)NN * HID;           // [N, 16]

    // zero the two scatter accumulators (s1|s2 are contiguous)
    {
        int zn = 2 * NN * HID;
        GCN_zero_f32<<<(zn + 255) / 256, 256, 0, stream>>>(s1, zn);
    }
    // K1: y1 = X @ W1 via WMMA f32 16x16x4, 32 rows per wave
    {
        int numSlabs = NN / 32;                  // 3125, exact
        GCN_gemm_x_w1<<<(numSlabs + 3) / 4, 128, 0, stream>>>(x, w1, y1);
    }
    // K2: s1 = A @ y1
    GCN_spmm_scatter<false><<<(EE + 255) / 256, 256, 0, stream>>>(
        esrc, edst, ew, y1, s1);
    // K3: s2 = A @ relu(s1)
    GCN_spmm_scatter<true><<<(EE + 255) / 256, 256, 0, stream>>>(
        esrc, edst, ew, s1, s2);
    // K4: out = log_softmax(s2 @ W2)
    GCN_head<<<(NN + 255) / 256, 256, 0, stream>>>(s2, w2, out);
}